// DeepONet_17952963297508
// MI455X (gfx1250) — compile-verified
//
#include <hip/hip_runtime.h>
#include <math.h>

// CDNA5 fp32 WMMA fragment types: A/B = 2 VGPRs, C/D = 8 VGPRs (wave32)
typedef __attribute__((ext_vector_type(2))) float v2f;
typedef __attribute__((ext_vector_type(8))) float v8f;

// ---------------- workspace layout (float offsets) ----------------
// total 4,293,632 floats = 16.4 MB
#define WS_UV      0          // [8][128]
#define WS_UMU     1024       // [8][128]
#define WS_CW      2048       // [8][128]  Uv*Umu*w
#define WS_DV      3072       // [8][3][128]  (1-h1^2),(1-h2^2),(1-h3^2) branch_v
#define WS_DMU     6144       // [8][3][128]  branch_mu
#define WS_GVA     9216       // [8][128][128] chain ping
#define WS_GVB     140288     // [8][128][128] chain pong
#define WS_GMA     271360
#define WS_GMB     402432
#define WS_DUV     533504     // [8][128][200] full Jacobian of branch_v
#define WS_DUMU    738304     // [8][128][100]
#define WS_T       840704     // [3][8][128][100] coefficient mats for out1/2/3
#define WS_H1      1147904    // [4096][128] trunk ping (value)
#define WS_TX1     1672192    //              d/dx tangent
#define WS_TY1     2196480    //              d/dy tangent
#define WS_H2      2720768    // trunk pong (final h4 lands here)
#define WS_TX2     3245056
#define WS_TY2     3769344

#define OUT1_OFF   4096
#define OUT4_OFF   1232896
#define OUT5_OFF   1236992

// ============ branch forward passes + chain seed + cw ============
__global__ __launch_bounds__(128)
void donet_fwd_branches(const float* __restrict__ vx, const float* __restrict__ vy,
                        const float* __restrict__ DT,
                        const float* __restrict__ W0v, const float* __restrict__ b0v,
                        const float* __restrict__ W1v, const float* __restrict__ b1v,
                        const float* __restrict__ W2v, const float* __restrict__ b2v,
                        const float* __restrict__ W3v, const float* __restrict__ b3v,
                        const float* __restrict__ W0m, const float* __restrict__ b0m,
                        const float* __restrict__ W1m, const float* __restrict__ b1m,
                        const float* __restrict__ W2m, const float* __restrict__ b2m,
                        const float* __restrict__ W3m, const float* __restrict__ b3m,
                        const float* __restrict__ linw,
                        float* __restrict__ ws)
{
    __shared__ float xin[200];
    __shared__ float hA[128];
    __shared__ float hB[128];
    const int b = blockIdx.x;
    const int j = threadIdx.x;

    // ---------------- branch_v (din=200, interleaved x0,y0,x1,y1,...) ----------------
    for (int p = j; p < 200; p += 128)
        xin[p] = (p & 1) ? vy[b * 100 + (p >> 1)] : vx[b * 100 + (p >> 1)];
    __syncthreads();

    float z = b0v[j];
    for (int p = 0; p < 200; ++p) z = fmaf(xin[p], W0v[p * 128 + j], z);
    float h = tanhf(z); hA[j] = h; ws[WS_DV + b * 384 + j] = 1.f - h * h;
    __syncthreads();

    z = b1v[j];
    for (int k = 0; k < 128; ++k) z = fmaf(hA[k], W1v[k * 128 + j], z);
    h = tanhf(z); hB[j] = h; ws[WS_DV + b * 384 + 128 + j] = 1.f - h * h;
    __syncthreads();

    z = b2v[j];
    for (int k = 0; k < 128; ++k) z = fmaf(hB[k], W2v[k * 128 + j], z);
    h = tanhf(z); hA[j] = h; ws[WS_DV + b * 384 + 256 + j] = 1.f - h * h;
    __syncthreads();

    z = b3v[j];
    for (int k = 0; k < 128; ++k) z = fmaf(hA[k], W3v[k * 128 + j], z);
    const float uv = tanhf(z);
    const float d4v = 1.f - uv * uv;
    ws[WS_UV + b * 128 + j] = uv;
    // chain seed: G1[j][k] = d4[j] * W3^T[j][k] = d4[j]*W3[k][j]
    for (int k = 0; k < 128; ++k)
        ws[WS_GVA + b * 16384 + j * 128 + k] = d4v * W3v[k * 128 + j];
    __syncthreads();

    // ---------------- branch_mu (din=100) ----------------
    for (int p = j; p < 100; p += 128) xin[p] = DT[b * 100 + p];
    __syncthreads();

    z = b0m[j];
    for (int p = 0; p < 100; ++p) z = fmaf(xin[p], W0m[p * 128 + j], z);
    h = tanhf(z); hA[j] = h; ws[WS_DMU + b * 384 + j] = 1.f - h * h;
    __syncthreads();

    z = b1m[j];
    for (int k = 0; k < 128; ++k) z = fmaf(hA[k], W1m[k * 128 + j], z);
    h = tanhf(z); hB[j] = h; ws[WS_DMU + b * 384 + 128 + j] = 1.f - h * h;
    __syncthreads();

    z = b2m[j];
    for (int k = 0; k < 128; ++k) z = fmaf(hB[k], W2m[k * 128 + j], z);
    h = tanhf(z); hA[j] = h; ws[WS_DMU + b * 384 + 256 + j] = 1.f - h * h;
    __syncthreads();

    z = b3m[j];
    for (int k = 0; k < 128; ++k) z = fmaf(hA[k], W3m[k * 128 + j], z);
    const float um = tanhf(z);
    const float d4m = 1.f - um * um;
    ws[WS_UMU + b * 128 + j] = um;
    for (int k = 0; k < 128; ++k)
        ws[WS_GMA + b * 16384 + j * 128 + k] = d4m * W3m[k * 128 + j];

    ws[WS_CW + b * 128 + j] = uv * um * linw[j];
}

// ============ Jacobian chain step:  C[b] = (A[b] col-scaled by s[b]) * W^T ============
// A: [128][128], W: [N rows][128] row-major (so B[k][n] = W[n*128+k]), M = K = 128.
// 8 waves per block, one 16x16 C tile per wave, V_WMMA_F32_16X16X4_F32 over K.
// Bounds handled branch-free: clamped index + 0/1 mask multiply (no EXEC churn).
__global__ __launch_bounds__(256)
void donet_chain_gemm(const float* __restrict__ A, int sA,
                      const float* __restrict__ s, int sS,
                      const float* __restrict__ W,
                      float* __restrict__ C, int sC, int ldc, int N, int tilesN)
{
    const int wave = threadIdx.x >> 5;
    const int lane = threadIdx.x & 31;
    const int tile = blockIdx.x * 8 + wave;
    const int tilesPerB = 8 * tilesN;           // tilesM = 8 (M=128)
    if (tile >= tilesPerB * 8) return;          // uniform per wave
    const int b   = tile / tilesPerB;
    const int rem = tile - b * tilesPerB;
    const int nt  = rem & (0);                  // placeholder (computed below)
    (void)nt;
    const int mt  = rem % 8;                    // m-tile fastest: waves in a block span m for shared B? no:
    const int ntile = rem / 8;                  // n-tile slow
    const int m0 = mt * 16;
    const int n0 = ntile * 16;

    const int half = lane >> 4;
    const int r    = lane & 15;
    const float* Ab = A + b * sA;
    const float* sb = s + b * sS;
    float*       Cb = C + b * sC;

    const int   n   = n0 + r;
    const bool  nok = (n < N);
    const int   ncl = nok ? n : 0;              // clamped column
    const float msk = nok ? 1.f : 0.f;          // 0/1 mask, folded into B values

    v8f acc = {0.f, 0.f, 0.f, 0.f, 0.f, 0.f, 0.f, 0.f};
    const int arow = (m0 + r) * 128;
    const int brow = ncl * 128;
    for (int k0 = 0; k0 < 128; k0 += 4) {
        const int ka = k0 + (half << 1);        // lanes 0-15: K=k0,k0+1 ; 16-31: K=k0+2,k0+3
        v2f af, bf;
        af.x = Ab[arow + ka]     * sb[ka];
        af.y = Ab[arow + ka + 1] * sb[ka + 1];
        bf.x = W[brow + ka]     * msk;          // W^T gather (transpose free), unconditional load
        bf.y = W[brow + ka + 1] * msk;
        acc = __builtin_amdgcn_wmma_f32_16x16x4_f32(false, af, false, bf,
                                                    (short)0, acc, false, false);
    }
    if (nok) {
#pragma unroll
        for (int v = 0; v < 8; ++v)
            Cb[(m0 + v + half * 8) * ldc + n] = acc[v];
    }
}

// ============ coefficient matrices T1/T2/T3 for out1/2/3 ============
__global__ void donet_coeffs(float* __restrict__ ws, const float* __restrict__ linw)
{
    const int idx = blockIdx.x * blockDim.x + threadIdx.x;   // 8*128*100
    if (idx >= 102400) return;
    const int s = idx % 100;
    const int j = (idx / 100) & 127;
    const int b = idx / 12800;
    const float w  = linw[j];
    const float gm = ws[WS_UMU + b * 128 + j] * w;  // Umu*w for vx/vy terms
    const float gv = ws[WS_UV  + b * 128 + j] * w;  // Uv*w  for mu term
    ws[WS_T +           b * 12800 + j * 100 + s] = ws[WS_DUV  + b * 25600 + j * 200 + 2 * s]     * gm;
    ws[WS_T + 102400  + b * 12800 + j * 100 + s] = ws[WS_DUV  + b * 25600 + j * 200 + 2 * s + 1] * gm;
    ws[WS_T + 204800  + b * 12800 + j * 100 + s] = ws[WS_DUMU + b * 12800 + j * 100 + s]         * gv;
}

// ============ trunk layer 0: din=2, elementwise (value + 2 tangents) ============
__global__ void donet_trunk0(const float* __restrict__ cx, const float* __restrict__ cy,
                             const float* __restrict__ W0, const float* __restrict__ b0,
                             float* __restrict__ ws)
{
    const int idx = blockIdx.x * blockDim.x + threadIdx.x;   // 4096*128
    if (idx >= 524288) return;
    const int q = idx >> 7;
    const int j = idx & 127;
    const float wx = W0[j], wy = W0[128 + j];
    const float z = fmaf(cx[q], wx, fmaf(cy[q], wy, b0[j]));
    const float h = tanhf(z);
    const float d = 1.f - h * h;
    ws[WS_H1  + idx] = h;
    ws[WS_TX1 + idx] = wx * d;
    ws[WS_TY1 + idx] = wy * d;
}

// ============ fused trunk layer: value + jacfwd tangents, shared B fragment ============
// [4096,128] x [128,128]; 8 waves/block, one 16x16 tile per wave, 3 WMMA accumulators.
// n-tile varies fastest within a block so the 8 waves share the same A rows (WGP$ reuse).
__global__ __launch_bounds__(256)
void donet_trunk_layer(const float* __restrict__ Hin, const float* __restrict__ Txin,
                       const float* __restrict__ Tyin,
                       const float* __restrict__ W, const float* __restrict__ bias,
                       float* __restrict__ Hout, float* __restrict__ Txout,
                       float* __restrict__ Tyout)
{
    const int wave = threadIdx.x >> 5;
    const int lane = threadIdx.x & 31;
    const int tile = blockIdx.x * 8 + wave;     // 2048 tiles = 256 m-tiles x 8 n-tiles
    const int m0   = (tile >> 3) * 16;
    const int n0   = (tile & 7) * 16;
    const int half = lane >> 4;
    const int r    = lane & 15;
    const int arow = (m0 + r) * 128;
    const int col  = n0 + r;

    v8f cz = {0.f,0.f,0.f,0.f,0.f,0.f,0.f,0.f};
    v8f cx = {0.f,0.f,0.f,0.f,0.f,0.f,0.f,0.f};
    v8f cy = {0.f,0.f,0.f,0.f,0.f,0.f,0.f,0.f};
    for (int k0 = 0; k0 < 128; k0 += 4) {
        const int ka = k0 + (half << 1);
        v2f az, ax, ay, bf;
        az.x = Hin [arow + ka];  az.y = Hin [arow + ka + 1];
        ax.x = Txin[arow + ka];  ax.y = Txin[arow + ka + 1];
        ay.x = Tyin[arow + ka];  ay.y = Tyin[arow + ka + 1];
        bf.x = W[ka * 128 + col];       bf.y = W[(ka + 1) * 128 + col];
        cz = __builtin_amdgcn_wmma_f32_16x16x4_f32(false, az, false, bf, (short)0, cz, false, false);
        cx = __builtin_amdgcn_wmma_f32_16x16x4_f32(false, ax, false, bf, (short)0, cx, false, false);
        cy = __builtin_amdgcn_wmma_f32_16x16x4_f32(false, ay, false, bf, (short)0, cy, false, false);
    }
    const float bv = bias[col];
#pragma unroll
    for (int v = 0; v < 8; ++v) {
        const int orow = m0 + v + half * 8;
        const float h = tanhf(cz[v] + bv);
        const float d = 1.f - h * h;
        Hout [orow * 128 + col] = h;
        Txout[orow * 128 + col] = cx[v] * d;
        Tyout[orow * 128 + col] = cy[v] * d;
    }
}

// ============ out1/2/3:  per batch [512,128] x [128,100] ============
// 5376 tiles = (b,t:24) x (mt:32) x (nt:7); 8 waves/block; branch-free N guard.
__global__ __launch_bounds__(256)
void donet_gemm_out(const float* __restrict__ ws, float* __restrict__ out)
{
    const int wave = threadIdx.x >> 5;
    const int lane = threadIdx.x & 31;
    const int tile = blockIdx.x * 8 + wave;
    const int zb   = tile / 224;                 // 32*7 tiles per (b,t)
    const int rem  = tile - zb * 224;
    const int nt   = rem % 7;
    const int mt   = rem / 7;
    const int b    = zb / 3;
    const int t    = zb - 3 * b;
    const int m0   = mt * 16;
    const int n0   = nt * 16;
    const int half = lane >> 4;
    const int r    = lane & 15;

    const float* A  = ws + WS_H2 + b * 512 * 128;              // trunk h4 for this batch
    const float* T  = ws + WS_T + (t * 8 + b) * 12800;         // [128][100]
    float*       O  = out + OUT1_OFF + t * 409600 + b * 51200; // [512][100]
    const int   n   = n0 + r;
    const bool  nok = (n < 100);
    const int   ncl = nok ? n : 0;
    const float msk = nok ? 1.f : 0.f;
    const int arow  = (m0 + r) * 128;

    v8f acc = {0.f,0.f,0.f,0.f,0.f,0.f,0.f,0.f};
    for (int k0 = 0; k0 < 128; k0 += 4) {
        const int ka = k0 + (half << 1);
        v2f af, bf;
        af.x = A[arow + ka];
        af.y = A[arow + ka + 1];
        bf.x = T[ka * 100 + ncl]       * msk;   // unconditional loads, mask folded in
        bf.y = T[(ka + 1) * 100 + ncl] * msk;
        acc = __builtin_amdgcn_wmma_f32_16x16x4_f32(false, af, false, bf,
                                                    (short)0, acc, false, false);
    }
    if (nok) {
#pragma unroll
        for (int v = 0; v < 8; ++v)
            O[(m0 + v + half * 8) * 100 + n] = acc[v];
    }
}

// ============ out0 / out4 / out5: per-point J reduction ============
__global__ void donet_out045(const float* __restrict__ ws, float* __restrict__ out)
{
    const int q = blockIdx.x * blockDim.x + threadIdx.x;
    if (q >= 4096) return;
    const int b = q >> 9;
    const float* cw = ws + WS_CW  + b * 128;
    const float* H  = ws + WS_H2  + q * 128;
    const float* Tx = ws + WS_TX2 + q * 128;
    const float* Ty = ws + WS_TY2 + q * 128;
    float a0 = 0.f, a4 = 0.f, a5 = 0.f;
    for (int j = 0; j < 128; ++j) {
        const float c = cw[j];
        a0 = fmaf(H[j],  c, a0);
        a4 = fmaf(Tx[j], c, a4);
        a5 = fmaf(Ty[j], c, a5);
    }
    out[q]            = a0;
    out[OUT4_OFF + q] = a4;
    out[OUT5_OFF + q] = a5;
}

extern "C" void kernel_launch(void* const* d_in, const int* in_sizes, int n_in,
                              void* d_out, int out_size, void* d_ws, size_t ws_size,
                              hipStream_t stream) {
    const float* vx  = (const float*)d_in[0];
    const float* vy  = (const float*)d_in[1];
    const float* DT  = (const float*)d_in[2];
    const float* cox = (const float*)d_in[3];
    const float* coy = (const float*)d_in[4];
    const float* W0v = (const float*)d_in[5];  const float* b0v = (const float*)d_in[6];
    const float* W1v = (const float*)d_in[7];  const float* b1v = (const float*)d_in[8];
    const float* W2v = (const float*)d_in[9];  const float* b2v = (const float*)d_in[10];
    const float* W3v = (const float*)d_in[11]; const float* b3v = (const float*)d_in[12];
    const float* W0m = (const float*)d_in[13]; const float* b0m = (const float*)d_in[14];
    const float* W1m = (const float*)d_in[15]; const float* b1m = (const float*)d_in[16];
    const float* W2m = (const float*)d_in[17]; const float* b2m = (const float*)d_in[18];
    const float* W3m = (const float*)d_in[19]; const float* b3m = (const float*)d_in[20];
    const float* W0t = (const float*)d_in[21]; const float* b0t = (const float*)d_in[22];
    const float* W1t = (const float*)d_in[23]; const float* b1t = (const float*)d_in[24];
    const float* W2t = (const float*)d_in[25]; const float* b2t = (const float*)d_in[26];
    const float* W3t = (const float*)d_in[27]; const float* b3t = (const float*)d_in[28];
    const float* lnw = (const float*)d_in[29];
    float* out = (float*)d_out;
    float* ws  = (float*)d_ws;   // needs ~16.4 MB

    // 1) branch forward passes + chain seeds + cw
    donet_fwd_branches<<<8, 128, 0, stream>>>(vx, vy, DT,
        W0v, b0v, W1v, b1v, W2v, b2v, W3v, b3v,
        W0m, b0m, W1m, b1m, W2m, b2m, W3m, b3m, lnw, ws);

    // 2) branch_v Jacobian chain: G1 -> G2 -> G3 -> dUv [128,200]
    //    tiles = 8(m) * tilesN * 8(batch); 8 waves/block
    donet_chain_gemm<<<64,  256, 0, stream>>>(ws + WS_GVA, 16384, ws + WS_DV + 256, 384, W2v, ws + WS_GVB, 16384, 128, 128, 8);
    donet_chain_gemm<<<64,  256, 0, stream>>>(ws + WS_GVB, 16384, ws + WS_DV + 128, 384, W1v, ws + WS_GVA, 16384, 128, 128, 8);
    donet_chain_gemm<<<104, 256, 0, stream>>>(ws + WS_GVA, 16384, ws + WS_DV,       384, W0v, ws + WS_DUV, 25600, 200, 200, 13);

    // 3) branch_mu Jacobian chain -> dUmu [128,100]
    donet_chain_gemm<<<64, 256, 0, stream>>>(ws + WS_GMA, 16384, ws + WS_DMU + 256, 384, W2m, ws + WS_GMB, 16384, 128, 128, 8);
    donet_chain_gemm<<<64, 256, 0, stream>>>(ws + WS_GMB, 16384, ws + WS_DMU + 128, 384, W1m, ws + WS_GMA, 16384, 128, 128, 8);
    donet_chain_gemm<<<56, 256, 0, stream>>>(ws + WS_GMA, 16384, ws + WS_DMU,       384, W0m, ws + WS_DUMU, 12800, 100, 100, 7);

    // 4) coefficient matrices
    donet_coeffs<<<400, 256, 0, stream>>>(ws, lnw);

    // 5) trunk: layer0 elementwise, then 3 fused WMMA layers (ping-pong; final set = H2/TX2/TY2)
    donet_trunk0<<<2048, 256, 0, stream>>>(cox, coy, W0t, b0t, ws);
    donet_trunk_layer<<<256, 256, 0, stream>>>(ws + WS_H1, ws + WS_TX1, ws + WS_TY1, W1t, b1t, ws + WS_H2, ws + WS_TX2, ws + WS_TY2);
    donet_trunk_layer<<<256, 256, 0, stream>>>(ws + WS_H2, ws + WS_TX2, ws + WS_TY2, W2t, b2t, ws + WS_H1, ws + WS_TX1, ws + WS_TY1);
    donet_trunk_layer<<<256, 256, 0, stream>>>(ws + WS_H1, ws + WS_TX1, ws + WS_TY1, W3t, b3t, ws + WS_H2, ws + WS_TX2, ws + WS_TY2);

    // 6) out1/out2/out3 batched GEMMs (5376 tiles / 8 waves)
    donet_gemm_out<<<672, 256, 0, stream>>>(ws, out);

    // 7) out0/out4/out5 reductions
    donet_out045<<<16, 256, 0, stream>>>(ws, out);
}